// L21_res_5763846111802
// MI455X (gfx1250) — compile-verified
//
#include <hip/hip_runtime.h>

typedef __attribute__((ext_vector_type(2))) float v2f;
typedef __attribute__((ext_vector_type(8))) float v8f;

__device__ __forceinline__ float wave_reduce_sum(float v) {
#pragma unroll
    for (int m = 16; m >= 1; m >>= 1)
        v += __shfl_xor(v, m, 32);
    return v;
}

// Pass 1: each block writes one fp32 partial sum of (a-b)^2 into partials[blockIdx.x].
// Main accumulation runs through v_wmma_f32_16x16x4_f32 with B = ones:
// D = A*1 + C accumulates row-sums of the 16x4 A tile holding squared diffs.
__global__ __launch_bounds__(256) void ssd_partial_kernel(
    const float* __restrict__ a, const float* __restrict__ b,
    float* __restrict__ partials, int n) {
    const int tid   = threadIdx.x;
    const int gid   = blockIdx.x * blockDim.x + tid;
    const int total = gridDim.x * blockDim.x;

    const int n4         = n >> 2;               // float4 count
    const int full_steps = n4 / total;           // wave-uniform trip count (EXEC all-1s for WMMA)
    const int rem4       = n4 - full_steps * total;

    const float4* __restrict__ a4 = (const float4*)a;
    const float4* __restrict__ b4 = (const float4*)b;

    v8f c = {0.f, 0.f, 0.f, 0.f, 0.f, 0.f, 0.f, 0.f};
    const v2f ones = {1.0f, 1.0f};

#pragma unroll 2
    for (int s = 0; s < full_steps; ++s) {
        const int i = gid + s * total;
        const float4 x = a4[i];                  // global_load_b128
        const float4 y = b4[i];
        const float d0 = x.x - y.x, d1 = x.y - y.y;
        const float d2 = x.z - y.z, d3 = x.w - y.w;
        v2f a01 = {d0 * d0, d1 * d1};
        v2f a23 = {d2 * d2, d3 * d3};
        // 8 args: (neg_a, A, neg_b, B, c_mod, C, reuse_a, reuse_b)
        c = __builtin_amdgcn_wmma_f32_16x16x4_f32(false, a01, false, ones,
                                                  (short)0, c, false, false);
        c = __builtin_amdgcn_wmma_f32_16x16x4_f32(false, a23, false, ones,
                                                  (short)0, c, false, false);
    }

    // Ragged remainder: scalar path (may diverge -> keep out of WMMA).
    float t = 0.0f;
    if (gid < rem4) {
        const int i = full_steps * total + gid;
        const float4 x = a4[i];
        const float4 y = b4[i];
        const float d0 = x.x - y.x, d1 = x.y - y.y;
        const float d2 = x.z - y.z, d3 = x.w - y.w;
        t = d0 * d0 + d1 * d1 + d2 * d2 + d3 * d3;
    }
    for (int i = (n4 << 2) + gid; i < n; i += total) {  // n % 4 stragglers
        const float d = a[i] - b[i];
        t += d * d;
    }

    // Every D column accumulated the same row-sums -> all-lane sum is 16x the
    // wave partial; the /16 is an exact power-of-two scale.
    const float s8 = c[0] + c[1] + c[2] + c[3] + c[4] + c[5] + c[6] + c[7];
    const float wsum = wave_reduce_sum(s8) * 0.0625f + wave_reduce_sum(t);

    __shared__ float red[8];
    const int lane = tid & 31;
    const int wid  = tid >> 5;
    if (lane == 0) red[wid] = wsum;
    __syncthreads();
    if (tid == 0) {
        float bs = 0.f;
#pragma unroll
        for (int w = 0; w < 8; ++w) bs += red[w];
        partials[blockIdx.x] = bs;
    }
}

// Pass 2: deterministic single-block reduction of the per-block partials.
__global__ __launch_bounds__(256) void ssd_finalize_kernel(
    const float* __restrict__ partials, int nblocks,
    float* __restrict__ out, float scale) {
    __shared__ float red[256];
    float s = 0.f;
    for (int i = threadIdx.x; i < nblocks; i += 256) s += partials[i];
    red[threadIdx.x] = s;
    __syncthreads();
#pragma unroll
    for (int off = 128; off >= 1; off >>= 1) {
        if (threadIdx.x < off) red[threadIdx.x] += red[threadIdx.x + off];
        __syncthreads();
    }
    if (threadIdx.x == 0) out[0] = red[0] * scale;
}

extern "C" void kernel_launch(void* const* d_in, const int* in_sizes, int n_in,
                              void* d_out, int out_size, void* d_ws, size_t ws_size,
                              hipStream_t stream) {
    const float* img1 = (const float*)d_in[0];
    const float* img2 = (const float*)d_in[1];
    float* out        = (float*)d_out;
    float* partials   = (float*)d_ws;            // 1024 * 4 B = 4 KB scratch
    const int n       = in_sizes[0];             // 12,582,912 elements

    constexpr int BLOCKS  = 1024;
    constexpr int THREADS = 256;                 // 8 wave32 waves per block

    ssd_partial_kernel<<<BLOCKS, THREADS, 0, stream>>>(img1, img2, partials, n);
    // out = 2 * total_sum / N  (partials already carry the exact /16 WMMA fix-up)
    ssd_finalize_kernel<<<1, THREADS, 0, stream>>>(partials, BLOCKS, out,
                                                   2.0f / (float)n);
}